// CustomModelCTCLoss_48430051230470
// MI455X (gfx1250) — compile-verified
//
#include <hip/hip_runtime.h>
#include <stdint.h>

#ifndef __has_builtin
#define __has_builtin(x) 0
#endif

#define B_ 64
#define T_ 2000
#define V_ 29
#define L_ 200
#define S_ (2 * L_ + 1)      /* 401 */
#define BLANK_ 28
#define NEG_ (-1e30f)
#define EPS_ (1e-7f)
#define CHUNK_ 80            /* 25 chunks cover T=2000 exactly */
#define NTHREADS_ 416        /* 13 wave32 waves, >= S_ */
#define ITEMS_ ((CHUNK_ * V_ * 4) / 16)   /* 580 16-byte packets per chunk */

#if __has_builtin(__builtin_amdgcn_global_load_async_to_lds_b128)
#define CTC_ASYNC 1
#endif

#ifdef CTC_ASYNC
typedef int v4i_ __attribute__((vector_size(16)));
typedef __attribute__((address_space(1))) v4i_* g_v4i_ptr;
typedef __attribute__((address_space(3))) v4i_* l_v4i_ptr;
#endif

__device__ __forceinline__ void async_copy16(const float* g, float* l) {
#ifdef CTC_ASYNC
  __builtin_amdgcn_global_load_async_to_lds_b128(
      (g_v4i_ptr)g, (l_v4i_ptr)l, /*offset=*/0, /*cpol=*/0);
#else
  *(float4*)l = *(const float4*)g;   // synchronous fallback
#endif
}

__device__ __forceinline__ void wait_async_le2() {
#ifdef CTC_ASYNC
#if __has_builtin(__builtin_amdgcn_s_wait_asynccnt)
  __builtin_amdgcn_s_wait_asynccnt(2);
#else
  asm volatile("s_wait_asynccnt 2" ::: "memory");
#endif
#endif
}

__device__ __forceinline__ void wait_async_le0() {
#ifdef CTC_ASYNC
#if __has_builtin(__builtin_amdgcn_s_wait_asynccnt)
  __builtin_amdgcn_s_wait_asynccnt(0);
#else
  asm volatile("s_wait_asynccnt 0" ::: "memory");
#endif
#endif
}

__global__ __launch_bounds__(NTHREADS_, 1) void ctc_loss_kernel(
    const float* __restrict__ logits,       // [B,T,V]
    const int* __restrict__ labels,         // [B,L]
    const int* __restrict__ input_length,   // [B]
    const int* __restrict__ label_length,   // [B]
    const int* __restrict__ mts_ptr,        // scalar
    float* __restrict__ out)                // [B,1]
{
  __shared__ __align__(16) float lp[2][CHUNK_ * V_];  // raw logits -> log-probs, in place
  __shared__ float alphaA[S_ + 2];                    // [0..1] = NEG guard band
  __shared__ float alphaB[S_ + 2];
  __shared__ float rowm[CHUNK_], rowlz[CHUNK_];

  const int b   = blockIdx.x;
  const int tid = threadIdx.x;

  // compute_length_after_conv: floor(il * T / max_time_steps), as f32 like the reference
  const int mts = mts_ptr[0];
  int ctc_len = (int)floorf((float)(input_length[b] * T_) / (float)mts);
  if (ctc_len < 1)  ctc_len = 1;
  if (ctc_len > T_) ctc_len = T_;

  // Per-state class id and skip-transition flag (ext label lattice)
  int  cls  = BLANK_;
  bool skip = false;
  if (tid < S_ && (tid & 1)) {
    const int k = tid >> 1;
    cls = labels[b * L_ + k];
    if (k >= 1) {
      const int prev = labels[b * L_ + k - 1];
      skip = (cls != BLANK_) && (cls != prev);
    }
  }
  if (tid < 2) { alphaA[tid] = NEG_; alphaB[tid] = NEG_; }

  const float* gbase  = logits + (size_t)b * T_ * V_;
  const int nchunks   = (ctc_len + CHUNK_ - 1) / CHUNK_;

  // Prefetch chunk 0 (async): 580 x 16B, 2 packets per lane.
  {
    int i = tid;
    async_copy16(gbase + i * 4, lp[0] + i * 4);
    i = tid + NTHREADS_; if (i >= ITEMS_) i = ITEMS_ - 1;   // duplicate tail, benign
    async_copy16(gbase + i * 4, lp[0] + i * 4);
  }

  float* cur = alphaA;
  float* nxt = alphaB;
  float  areg = NEG_;            // this thread's own alpha[s], carried in a register

  for (int c = 0; c < nchunks; ++c) {
    const int base = c * CHUNK_;
    float* buf = lp[c & 1];

    if (c + 1 < nchunks) {
      // Overlap: issue next chunk, then retire the current one (in-order per wave).
      const float* g = gbase + (size_t)(base + CHUNK_) * V_;
      float* l2 = lp[(c + 1) & 1];
      int i = tid;
      async_copy16(g + i * 4, l2 + i * 4);
      i = tid + NTHREADS_; if (i >= ITEMS_) i = ITEMS_ - 1;
      async_copy16(g + i * 4, l2 + i * 4);
      wait_async_le2();
    } else {
      wait_async_le0();
    }
    __syncthreads();

    // logits row -> log(softmax + eps), in place in LDS
    if (tid < CHUNK_) {
      const float* r = buf + tid * V_;
      float m = r[0];
#pragma unroll
      for (int v = 1; v < V_; ++v) m = fmaxf(m, r[v]);
      float z = 0.f;
#pragma unroll
      for (int v = 0; v < V_; ++v) z += __expf(r[v] - m);
      rowm[tid]  = m;
      rowlz[tid] = __logf(z);
    }
    __syncthreads();
    for (int i = tid; i < CHUNK_ * V_; i += NTHREADS_) {
      const int r = i / V_;
      const float v = buf[i] - rowm[r] - rowlz[r];
      buf[i] = __logf(__expf(v) + EPS_);
    }
    __syncthreads();

    int t0 = base;
    if (c == 0) {
      // alpha(t=0): states 0,1 get lp[0][ext[s]], rest NEG
      if (tid < S_) {
        areg = (tid < 2) ? buf[cls] : NEG_;
        cur[tid + 2] = areg;
      }
      __syncthreads();
      t0 = 1;
    }
    int t1 = base + CHUNK_; if (t1 > ctc_len) t1 = ctc_len;

    for (int t = t0; t < t1; ++t) {
      const float* rowlp = buf + (t - base) * V_;
      if (tid < S_) {
        const float a1 = cur[tid + 1];              // alpha[s-1] (guard band at s<1)
        const float a2 = skip ? cur[tid] : NEG_;    // alpha[s-2]
        const float lpv = rowlp[cls];
        const float m  = fmaxf(areg, fmaxf(a1, a2));
        const float s  = __expf(areg - m) + __expf(a1 - m) + __expf(a2 - m);
        areg = m + __logf(s) + lpv;
        nxt[tid + 2] = areg;
      }
      float* tmp = cur; cur = nxt; nxt = tmp;       // ping-pong: one barrier suffices
      __syncthreads();
    }
  }

  if (tid == 0) {
    int ll = label_length[b];
    if (ll < 0) ll = 0; if (ll > L_) ll = L_;
    const int i1 = 2 * ll;
    int i2 = i1 - 1; if (i2 < 0) i2 = 0;
    const float a1 = cur[i1 + 2];
    const float a2 = cur[i2 + 2];
    const float m  = fmaxf(a1, a2);
    out[b] = -(m + __logf(__expf(a1 - m) + __expf(a2 - m)));
  }
}

extern "C" void kernel_launch(void* const* d_in, const int* in_sizes, int n_in,
                              void* d_out, int out_size, void* d_ws, size_t ws_size,
                              hipStream_t stream) {
  const float* logits       = (const float*)d_in[0];
  const int*   labels       = (const int*)d_in[1];
  const int*   input_length = (const int*)d_in[2];
  const int*   label_length = (const int*)d_in[3];
  const int*   mts          = (const int*)d_in[4];
  ctc_loss_kernel<<<dim3(B_), dim3(NTHREADS_), 0, stream>>>(
      logits, labels, input_length, label_length, mts, (float*)d_out);
}